// GAT_21569325761083
// MI455X (gfx1250) — compile-verified
//
#include <hip/hip_runtime.h>
#include <math.h>

typedef __attribute__((ext_vector_type(2))) float v2f;
typedef __attribute__((ext_vector_type(8))) float v8f;

#define N_NODES 4096
#define DIN     512
#define FDIM    64
#define NHEAD   8
#define DOUT    16
#define GAT_ALPHA 0.2f

__device__ __forceinline__ float lrelu(float x) { return x > 0.f ? x : GAT_ALPHA * x; }

// ---------------------------------------------------------------------------
// Kernel A: Wh[h] = X @ W[h]   ([4096,512] @ [512,64]) via V_WMMA_F32_16X16X4_F32
// grid = (N/16, H), block = 128 (4 waves, one 16-wide f-tile per wave)
// ---------------------------------------------------------------------------
__global__ __launch_bounds__(128) void gat_gemm_xw(const float* __restrict__ X,
                                                   const float* __restrict__ W,
                                                   float* __restrict__ Wh) {
  const int h    = blockIdx.y;
  const int n0   = blockIdx.x * 16;
  const int wave = threadIdx.x >> 5;
  const int lane = threadIdx.x & 31;
  const int hi   = lane >> 4;          // K-half select (0 -> K{0,1}, 1 -> K{2,3})
  const int l16  = lane & 15;
  const int f0   = wave * 16;

  const float* Wp   = W + (size_t)h * DIN * FDIM;
  const float* arow = X  + (size_t)(n0 + l16) * DIN + 2 * hi;       // A: row n0+l16
  const float* bcol = Wp + (size_t)(2 * hi) * FDIM + (f0 + l16);    // B: col f0+l16

  v8f c = {};
  #pragma unroll 4
  for (int k = 0; k < DIN; k += 4) {
    v2f a, b;
    a.x = arow[k];
    a.y = arow[k + 1];
    b.x = bcol[(size_t)k * FDIM];
    b.y = bcol[(size_t)(k + 1) * FDIM];
    c = __builtin_amdgcn_wmma_f32_16x16x4_f32(false, a, false, b, (short)0, c,
                                              false, false);
  }
  float* orow = Wh + ((size_t)h * N_NODES + n0 + 8 * hi) * FDIM + f0 + l16;
  #pragma unroll
  for (int j = 0; j < 8; ++j) orow[(size_t)j * FDIM] = c[j];
}

// ---------------------------------------------------------------------------
// Kernel B: f1[h,n] = Wh[h,n,:]·a1[h,:],  f2 likewise. One thread per (h,n).
// ---------------------------------------------------------------------------
__global__ void gat_f12(const float* __restrict__ Wh, const float* __restrict__ a1,
                        const float* __restrict__ a2, float* __restrict__ f1,
                        float* __restrict__ f2) {
  int idx = blockIdx.x * blockDim.x + threadIdx.x;   // idx = h*N + n
  if (idx >= NHEAD * N_NODES) return;
  int h = idx >> 12;
  const float* w  = Wh + (size_t)idx * FDIM;
  const float* v1 = a1 + h * FDIM;
  const float* v2 = a2 + h * FDIM;
  float s1 = 0.f, s2 = 0.f;
  #pragma unroll 8
  for (int f = 0; f < FDIM; ++f) { s1 += w[f] * v1[f]; s2 += w[f] * v2[f]; }
  f1[idx] = s1;
  f2[idx] = s2;
}

// ---------------------------------------------------------------------------
// Kernel C: per-row masked softmax max (layer 1).
// emax1[h,n] = lrelu(f1[h,n] + max_{m: adj[n,m]>0} f2[h,m])  (lrelu is monotone)
// One block per row n.
// ---------------------------------------------------------------------------
__global__ __launch_bounds__(256) void gat_rowmax1(const float* __restrict__ adj,
                                                   const float* __restrict__ f1,
                                                   const float* __restrict__ f2,
                                                   float* __restrict__ emax1) {
  const int n = blockIdx.x;
  const float* arow = adj + (size_t)n * N_NODES;
  float mx[NHEAD];
  #pragma unroll
  for (int h = 0; h < NHEAD; ++h) mx[h] = -3.4e38f;
  for (int m = threadIdx.x; m < N_NODES; m += 256) {
    if (arow[m] > 0.f) {
      #pragma unroll
      for (int h = 0; h < NHEAD; ++h)
        mx[h] = fmaxf(mx[h], f2[h * N_NODES + m]);
    }
  }
  __shared__ float smx[NHEAD][256];
  #pragma unroll
  for (int h = 0; h < NHEAD; ++h) smx[h][threadIdx.x] = mx[h];
  __syncthreads();
  for (int s = 128; s > 0; s >>= 1) {
    if (threadIdx.x < s) {
      #pragma unroll
      for (int h = 0; h < NHEAD; ++h)
        smx[h][threadIdx.x] = fmaxf(smx[h][threadIdx.x], smx[h][threadIdx.x + s]);
    }
    __syncthreads();
  }
  if (threadIdx.x < NHEAD) {
    int h = threadIdx.x;
    emax1[h * N_NODES + n] = lrelu(f1[h * N_NODES + n] + smx[h][0]);
  }
}

// ---------------------------------------------------------------------------
// Kernel D: layer-1 attention aggregation + ELU + concat.
// One block (512 threads) per row n. Phase 1 compacts edges of a 512-col chunk
// into LDS with per-head softmax weights; phase 2: thread (h,f) accumulates.
// ---------------------------------------------------------------------------
__global__ __launch_bounds__(512) void gat_attagg1(const float* __restrict__ adj,
                                                   const float* __restrict__ f1,
                                                   const float* __restrict__ f2,
                                                   const float* __restrict__ emax1,
                                                   const float* __restrict__ Wh,
                                                   float* __restrict__ cat) {
  const int n = blockIdx.x;
  __shared__ int   s_idx[512];
  __shared__ float s_w[NHEAD][512];
  __shared__ int   s_cnt;
  __shared__ float s_f1[NHEAD], s_em[NHEAD];
  if (threadIdx.x < NHEAD) {
    s_f1[threadIdx.x] = f1[threadIdx.x * N_NODES + n];
    s_em[threadIdx.x] = emax1[threadIdx.x * N_NODES + n];
  }
  const float* arow = adj + (size_t)n * N_NODES;
  const int h = threadIdx.x >> 6;
  const int f = threadIdx.x & 63;
  float acc = 0.f, dsum = 0.f;

  for (int c0 = 0; c0 < N_NODES; c0 += 512) {
    if (threadIdx.x == 0) s_cnt = 0;
    __syncthreads();                       // covers s_f1/s_em + cnt reset
    int m = c0 + threadIdx.x;
    if (arow[m] > 0.f) {
      int p = atomicAdd(&s_cnt, 1);
      s_idx[p] = m;
      #pragma unroll
      for (int hh = 0; hh < NHEAD; ++hh)
        s_w[hh][p] = expf(lrelu(s_f1[hh] + f2[hh * N_NODES + m]) - s_em[hh]);
    }
    __syncthreads();
    int cnt = s_cnt;
    for (int e = 0; e < cnt; ++e) {
      float w = s_w[h][e];
      acc  += w * Wh[((size_t)h * N_NODES + s_idx[e]) * FDIM + f];
      dsum += w;
    }
    __syncthreads();
  }
  float r = acc / dsum;
  r = r > 0.f ? r : (expf(r) - 1.f);       // ELU (concat=True heads)
  cat[(size_t)n * (NHEAD * FDIM) + h * FDIM + f] = r;
}

// ---------------------------------------------------------------------------
// Kernel E: Who = cat @ Wo   ([4096,512] @ [512,16]) via f32 WMMA. 1 wave/block.
// ---------------------------------------------------------------------------
__global__ __launch_bounds__(32) void gat_gemm_cat_wo(const float* __restrict__ cat,
                                                      const float* __restrict__ Wo,
                                                      float* __restrict__ Who) {
  const int n0  = blockIdx.x * 16;
  const int lane = threadIdx.x & 31;
  const int hi  = lane >> 4;
  const int l16 = lane & 15;
  const float* arow = cat + (size_t)(n0 + l16) * (NHEAD * FDIM) + 2 * hi;
  const float* bcol = Wo + (size_t)(2 * hi) * DOUT + l16;
  v8f c = {};
  #pragma unroll 4
  for (int k = 0; k < NHEAD * FDIM; k += 4) {
    v2f a, b;
    a.x = arow[k];
    a.y = arow[k + 1];
    b.x = bcol[(size_t)k * DOUT];
    b.y = bcol[(size_t)(k + 1) * DOUT];
    c = __builtin_amdgcn_wmma_f32_16x16x4_f32(false, a, false, b, (short)0, c,
                                              false, false);
  }
  float* orow = Who + (size_t)(n0 + 8 * hi) * DOUT + l16;
  #pragma unroll
  for (int j = 0; j < 8; ++j) orow[(size_t)j * DOUT] = c[j];
}

// ---------------------------------------------------------------------------
// Kernel F: go1[n] = Who[n,:]·ao1, go2[n] = Who[n,:]·ao2
// ---------------------------------------------------------------------------
__global__ void gat_go(const float* __restrict__ Who, const float* __restrict__ ao1,
                       const float* __restrict__ ao2, float* __restrict__ go1,
                       float* __restrict__ go2) {
  int n = blockIdx.x * blockDim.x + threadIdx.x;
  if (n >= N_NODES) return;
  float s1 = 0.f, s2 = 0.f;
  #pragma unroll
  for (int f = 0; f < DOUT; ++f) {
    float v = Who[(size_t)n * DOUT + f];
    s1 += v * ao1[f];
    s2 += v * ao2[f];
  }
  go1[n] = s1;
  go2[n] = s2;
}

// ---------------------------------------------------------------------------
// Kernel G: output-layer masked softmax max (single head).
// ---------------------------------------------------------------------------
__global__ __launch_bounds__(256) void gat_rowmax_o(const float* __restrict__ adj,
                                                    const float* __restrict__ go1,
                                                    const float* __restrict__ go2,
                                                    float* __restrict__ emaxo) {
  const int n = blockIdx.x;
  const float* arow = adj + (size_t)n * N_NODES;
  float mx = -3.4e38f;
  for (int m = threadIdx.x; m < N_NODES; m += 256)
    if (arow[m] > 0.f) mx = fmaxf(mx, go2[m]);
  __shared__ float smx[256];
  smx[threadIdx.x] = mx;
  __syncthreads();
  for (int s = 128; s > 0; s >>= 1) {
    if (threadIdx.x < s) smx[threadIdx.x] = fmaxf(smx[threadIdx.x], smx[threadIdx.x + s]);
    __syncthreads();
  }
  if (threadIdx.x == 0) emaxo[n] = lrelu(go1[n] + smx[0]);
}

// ---------------------------------------------------------------------------
// Kernel H: out = softmax(mask(eo)) @ Who  (single head, no ELU).
// ---------------------------------------------------------------------------
__global__ __launch_bounds__(256) void gat_attagg_o(const float* __restrict__ adj,
                                                    const float* __restrict__ go1,
                                                    const float* __restrict__ go2,
                                                    const float* __restrict__ emaxo,
                                                    const float* __restrict__ Who,
                                                    float* __restrict__ out) {
  const int n = blockIdx.x;
  __shared__ int   s_idx[256];
  __shared__ float s_w[256];
  __shared__ int   s_cnt;
  const float sf1 = go1[n];
  const float sem = emaxo[n];
  const float* arow = adj + (size_t)n * N_NODES;
  float acc = 0.f, dsum = 0.f;

  for (int c0 = 0; c0 < N_NODES; c0 += 256) {
    if (threadIdx.x == 0) s_cnt = 0;
    __syncthreads();
    int m = c0 + threadIdx.x;
    if (arow[m] > 0.f) {
      int p = atomicAdd(&s_cnt, 1);
      s_idx[p] = m;
      s_w[p] = expf(lrelu(sf1 + go2[m]) - sem);
    }
    __syncthreads();
    int cnt = s_cnt;
    if (threadIdx.x < DOUT) {
      for (int e = 0; e < cnt; ++e) {
        float w = s_w[e];
        acc  += w * Who[(size_t)s_idx[e] * DOUT + threadIdx.x];
        dsum += w;
      }
    }
    __syncthreads();
  }
  if (threadIdx.x < DOUT) out[(size_t)n * DOUT + threadIdx.x] = acc / dsum;
}

// ---------------------------------------------------------------------------
extern "C" void kernel_launch(void* const* d_in, const int* in_sizes, int n_in,
                              void* d_out, int out_size, void* d_ws, size_t ws_size,
                              hipStream_t stream) {
  (void)in_sizes; (void)n_in; (void)out_size; (void)ws_size;
  const float* adj = (const float*)d_in[0];
  const float* X   = (const float*)d_in[1];
  const float* W   = (const float*)d_in[2];
  const float* a1  = (const float*)d_in[3];
  const float* a2  = (const float*)d_in[4];
  const float* Wo  = (const float*)d_in[5];
  const float* ao1 = (const float*)d_in[6];
  const float* ao2 = (const float*)d_in[7];
  float* out = (float*)d_out;

  // workspace layout (floats)
  float* ws    = (float*)d_ws;
  float* Wh    = ws;                 // H*N*F      = 2,097,152
  float* f1    = Wh    + (size_t)NHEAD * N_NODES * FDIM;   //  32,768
  float* f2    = f1    + NHEAD * N_NODES;                  //  32,768
  float* emax1 = f2    + NHEAD * N_NODES;                  //  32,768
  float* cat   = emax1 + NHEAD * N_NODES;                  // 2,097,152
  float* Who   = cat   + (size_t)N_NODES * NHEAD * FDIM;   //  65,536
  float* go1   = Who   + N_NODES * DOUT;                   //   4,096
  float* go2   = go1   + N_NODES;                          //   4,096
  float* emaxo = go2   + N_NODES;                          //   4,096

  // Layer 1
  gat_gemm_xw<<<dim3(N_NODES / 16, NHEAD), 128, 0, stream>>>(X, W, Wh);
  gat_f12<<<(NHEAD * N_NODES + 255) / 256, 256, 0, stream>>>(Wh, a1, a2, f1, f2);
  gat_rowmax1<<<N_NODES, 256, 0, stream>>>(adj, f1, f2, emax1);
  gat_attagg1<<<N_NODES, 512, 0, stream>>>(adj, f1, f2, emax1, Wh, cat);

  // Output layer
  gat_gemm_cat_wo<<<N_NODES / 16, 32, 0, stream>>>(cat, Wo, Who);
  gat_go<<<(N_NODES + 255) / 256, 256, 0, stream>>>(Who, ao1, ao2, go1, go2);
  gat_rowmax_o<<<N_NODES, 256, 0, stream>>>(adj, go1, go2, emaxo);
  gat_attagg_o<<<N_NODES, 256, 0, stream>>>(adj, go1, go2, emaxo, Who, out);
}